// HiddenStateUserItem_31396210934416
// MI455X (gfx1250) — compile-verified
//
#include <hip/hip_runtime.h>

typedef float v2f __attribute__((ext_vector_type(2)));
typedef float v8f __attribute__((ext_vector_type(8)));

#define NUSER 5
#define NITEM 3
#define DIM   256
#define KNBR  200
#define JDIM  (KNBR * 2 * DIM)   // 102400
#define HID   256

// ---------------------------------------------------------------------------
// Kernel 1: gather concat_u_e [5, J] and concat_i_e [3, J] into workspace.
// One block per (side, node, neighbor); 128 threads copy 2 x 256 floats as float4.
// ---------------------------------------------------------------------------
__global__ void gather_kernel(const int* __restrict__ adj0, const int* __restrict__ adj1,
                              const int* __restrict__ adj2, const int* __restrict__ adj3,
                              const float* __restrict__ rev, const float* __restrict__ usr,
                              const float* __restrict__ itm,
                              float* __restrict__ Au, float* __restrict__ Ai) {
    int b = blockIdx.x;
    int t = threadIdx.x;                     // 0..127
    const float* src0;
    const float* src1;
    float* dst;
    if (b < NUSER * KNBR) {
        int v = b / KNBR, k = b % KNBR;
        src0 = rev + (size_t)adj0[v * KNBR + k] * DIM;
        src1 = itm + (size_t)adj1[v * KNBR + k] * DIM;
        dst  = Au + (size_t)v * JDIM + (size_t)k * 2 * DIM;
    } else {
        int p = b - NUSER * KNBR;
        int v = p / KNBR, k = p % KNBR;
        src0 = rev + (size_t)adj2[v * KNBR + k] * DIM;
        src1 = usr + (size_t)adj3[v * KNBR + k] * DIM;
        dst  = Ai + (size_t)v * JDIM + (size_t)k * 2 * DIM;
    }
    if (t < 64) {
        float4 x = ((const float4*)src0)[t];
        ((float4*)dst)[t] = x;
    } else {
        float4 x = ((const float4*)src1)[t - 64];
        ((float4*)(dst + DIM))[t - 64] = x;
    }
}

// ---------------------------------------------------------------------------
// Kernel 2: zero the split-K accumulators (G_u 16x256 + G_i 16x256, contiguous).
// Must run every launch (graph replays) since kernel 3 accumulates with atomics.
// ---------------------------------------------------------------------------
__global__ void zero_kernel(float* __restrict__ g) {
    g[blockIdx.x * blockDim.x + threadIdx.x] = 0.0f;
}

// ---------------------------------------------------------------------------
// Kernel 3: split-K streaming GEMM  G[V,256] = A[V,J] @ W[J,256]  via
// V_WMMA_F32_16X16X4_F32.  A is zero-padded to 16 rows in registers: all lanes
// load from a clamped (always-valid) row, invalid rows are zeroed with a
// v_cndmask select -- no EXEC manipulation in the hot loop.
// W loads are non-temporal (streamed exactly once); A stays cached.
// grid = (32 j-chunks, 16 n-tiles, 2 matrices), 8 waves per block.
// Each wave: 100 iterations of K=4, then f32 atomics into G (rows < V only).
// ---------------------------------------------------------------------------
__global__ void __launch_bounds__(256)
gemm_kernel(const float* __restrict__ Au, const float* __restrict__ Ai,
            const float* __restrict__ Wu, const float* __restrict__ Wi,
            float* __restrict__ Gu, float* __restrict__ Gi) {
    const float* A;
    const float* W;
    float* G;
    int V;
    if (blockIdx.z == 0) { A = Au; W = Wu; G = Gu; V = NUSER; }
    else                 { A = Ai; W = Wi; G = Gi; V = NITEM; }

    const int wave = threadIdx.x >> 5;
    const int lane = threadIdx.x & 31;
    const int m    = lane & 15;          // A row / B column within tile
    const int hi   = lane >> 4;          // lane-half selects K 0-1 vs 2-3
    const int n0   = blockIdx.y * 16;    // output column tile base

    const int wseg  = blockIdx.x * 8 + wave;          // 0..255 per (matrix, tile)
    const int steps = (JDIM / 4) / (32 * 8);          // 100 K=4 iterations per wave
    const int j0    = wseg * steps * 4;

    v8f c = {0.f, 0.f, 0.f, 0.f, 0.f, 0.f, 0.f, 0.f};

    // B pointer: row (j + 2*hi), column (n0 + m); rows stride HID floats.
    const float* Wp = W + (size_t)(j0 + 2 * hi) * HID + n0 + m;
    // A pointer: clamped row (always valid memory); invalid rows zeroed via select.
    const bool  arow = (m < V);
    const int   mrow = arow ? m : 0;
    const float* Ap  = A + (size_t)mrow * JDIM + j0 + 2 * hi;   // 8B aligned

    for (int s = 0; s < steps; ++s) {
        v2f a = *(const v2f*)Ap;                 // global_load_b64
        a.x = arow ? a.x : 0.0f;                 // v_cndmask, no branches
        a.y = arow ? a.y : 0.0f;

        v2f b;
        b.x = __builtin_nontemporal_load(Wp);        // W[j+2*hi  , n0+m], TH_NT
        b.y = __builtin_nontemporal_load(Wp + HID);  // W[j+2*hi+1, n0+m], TH_NT

        // stream-ahead prefetch of W (4 iterations = 16 rows = 16 KB ahead)
        __builtin_prefetch(Wp + 16 * HID, 0, 3);

        // D = A(16x4) * B(4x16) + C   (EXEC all-1s: no divergence in this loop)
        c = __builtin_amdgcn_wmma_f32_16x16x4_f32(
                false, a, false, b, (short)0, c, false, false);

        Wp += 4 * HID;
        Ap += 4;
    }

    // C/D layout: VGPR r -> M=r (lanes 0-15), M=8+r (lanes 16-31).
    // V <= 5, so only lanes 0-15 can hold valid rows.
    if (hi == 0) {
#pragma unroll
        for (int r = 0; r < 8; ++r) {
            if (r < V) atomicAdd(&G[(size_t)r * HID + n0 + m], c[r]);
        }
    }
}

// ---------------------------------------------------------------------------
// Kernel 4: epilogue.  4096 threads = out_size.
//   h <  256 : concat_*_weights @ embedding
//   h >= 256 : relu( softmax(E E^T / 16)[u,:] @ G )
// ---------------------------------------------------------------------------
__global__ void epilogue_kernel(const float* __restrict__ U, const float* __restrict__ Ie,
                                const float* __restrict__ Gu, const float* __restrict__ Gi,
                                const float* __restrict__ CWu, const float* __restrict__ CWi,
                                float* __restrict__ out) {
    int tid = blockIdx.x * blockDim.x + threadIdx.x;   // 0..4095
    int V, base, t;
    const float* E;
    const float* G;
    const float* CW;
    if (tid < NUSER * 512) { V = NUSER; E = U;  G = Gu; CW = CWu; base = 0;           t = tid; }
    else                   { V = NITEM; E = Ie; G = Gi; CW = CWi; base = NUSER * 512; t = tid - NUSER * 512; }

    int u = t / 512;
    int h = t % 512;
    float val;
    if (h < 256) {
        float acc = 0.f;
        for (int v = 0; v < V; ++v) acc += CW[u * V + v] * E[(size_t)v * DIM + h];
        val = acc;
    } else {
        int hh = h - 256;
        float s[NUSER];
        float mx = -3.4e38f;
        for (int v = 0; v < V; ++v) {
            float d = 0.f;
            for (int q = 0; q < DIM; ++q) d += E[(size_t)u * DIM + q] * E[(size_t)v * DIM + q];
            s[v] = d * (1.0f / 16.0f);                 // / sqrt(256)
            mx = fmaxf(mx, s[v]);
        }
        float sum = 0.f;
        for (int v = 0; v < V; ++v) { s[v] = __expf(s[v] - mx); sum += s[v]; }
        float acc = 0.f;
        for (int v = 0; v < V; ++v) acc += s[v] * G[(size_t)v * HID + hh];
        val = fmaxf(acc / sum, 0.0f);                  // relu
    }
    out[base + t] = val;
}

// ---------------------------------------------------------------------------
extern "C" void kernel_launch(void* const* d_in, const int* in_sizes, int n_in,
                              void* d_out, int out_size, void* d_ws, size_t ws_size,
                              hipStream_t stream) {
    const int*   adj0 = (const int*)d_in[0];
    const int*   adj1 = (const int*)d_in[1];
    const int*   adj2 = (const int*)d_in[2];
    const int*   adj3 = (const int*)d_in[3];
    const float* rev  = (const float*)d_in[4];
    const float* usr  = (const float*)d_in[5];
    const float* itm  = (const float*)d_in[6];
    const float* Wu   = (const float*)d_in[7];
    const float* Wi   = (const float*)d_in[8];
    const float* CWu  = (const float*)d_in[9];
    const float* CWi  = (const float*)d_in[10];

    char* ws = (char*)d_ws;
    float* Au = (float*)ws;                                          // 5*J floats
    float* Ai = (float*)(ws + sizeof(float) * (size_t)NUSER * JDIM); // 3*J floats
    float* Gu = (float*)(ws + sizeof(float) * (size_t)8 * JDIM);     // 16*256
    float* Gi = Gu + 16 * HID;                                       // 16*256

    hipLaunchKernelGGL(gather_kernel, dim3((NUSER + NITEM) * KNBR), dim3(128), 0, stream,
                       adj0, adj1, adj2, adj3, rev, usr, itm, Au, Ai);
    hipLaunchKernelGGL(zero_kernel, dim3(32), dim3(256), 0, stream, Gu);  // Gu+Gi contiguous
    hipLaunchKernelGGL(gemm_kernel, dim3(32, 16, 2), dim3(256), 0, stream,
                       Au, Ai, Wu, Wi, Gu, Gi);
    hipLaunchKernelGGL(epilogue_kernel, dim3(16), dim3(256), 0, stream,
                       usr, itm, Gu, Gi, CWu, CWi, (float*)d_out);
}